// Macenko_83777632075857
// MI455X (gfx1250) — compile-verified
//
#include <hip/hip_runtime.h>
#include <math.h>

// ---------------- constants ----------------
#define IO_VAL 240.0f
#define BETA_V 0.15f
#define PI_F   3.14159265358979323846f
#define NB     131072        // histogram bins (phi and each C row)
#define GRID1  1024          // blocks for the stats pass

typedef __attribute__((ext_vector_type(2))) float v2f;
typedef __attribute__((ext_vector_type(8))) float v8f;

// order-preserving float <-> uint key (for deterministic atomic min/max)
__device__ __forceinline__ unsigned fkey(float f) {
    unsigned b = __float_as_uint(f);
    return (b & 0x80000000u) ? ~b : (b | 0x80000000u);
}
__device__ __forceinline__ float fdec(unsigned k) {
    unsigned b = (k & 0x80000000u) ? (k & 0x7FFFFFFFu) : ~k;
    return __uint_as_float(b);
}

// ---------------- init: zero histograms, seed min/max keys ----------------
__global__ void mk_init(unsigned* __restrict__ hists, unsigned* __restrict__ mm, int nHist) {
    int i = blockIdx.x * blockDim.x + threadIdx.x;
    if (i < nHist) hists[i] = 0u;
    if (i == 0) { mm[0] = 0xFFFFFFFFu; mm[1] = 0u; mm[2] = 0xFFFFFFFFu; mm[3] = 0u; }
}

// ---------------- pass 1: OD + WMMA moment accumulation ----------------
// Per pixel v = (m*od0, m*od1, m*od2, m).  One f32 16x16x4 WMMA per 4 pixels
// accumulates D += A * A  with A[m][k] = v_k[m]; rows/cols 0..2 = sum od od^T,
// row 3 = sum od, [3][3] = n.  A-layout (32-bit 16x4): lane L -> row m=L%16,
// VGPR0 holds K=2h, VGPR1 K=2h+1, h=L/16; B 4x16 mirrors it, so a==b per lane.
__global__ __launch_bounds__(256) void mk_od_stats(
        const float* __restrict__ img, float* __restrict__ od,
        float* __restrict__ partials, int N, int perBlock) {
    __shared__ float tile4[256 * 4];
    __shared__ float wsum[8 * 16];
    const int tid  = threadIdx.x;
    const int lane = tid & 31;
    const int wv   = tid >> 5;
    const int mrow = lane & 15;
    const int half = lane >> 4;
    const int blockStart = blockIdx.x * perBlock;

    v8f acc = {};
    for (int base = 0; base < perBlock; base += 256) {
        int p = blockStart + base + tid;
        float m = 0.f, o0 = 0.f, o1 = 0.f, o2 = 0.f;
        if (p < N) {
            float r = img[p], g = img[N + p], b = img[2 * N + p];
            o0 = -logf((floorf(r * 255.f) + 1.f) / IO_VAL);
            o1 = -logf((floorf(g * 255.f) + 1.f) / IO_VAL);
            o2 = -logf((floorf(b * 255.f) + 1.f) / IO_VAL);
            od[p] = o0; od[N + p] = o1; od[2 * N + p] = o2;
            m = (o0 >= BETA_V && o1 >= BETA_V && o2 >= BETA_V) ? 1.f : 0.f;
        }
        tile4[tid * 4 + 0] = o0 * m;
        tile4[tid * 4 + 1] = o1 * m;
        tile4[tid * 4 + 2] = o2 * m;
        tile4[tid * 4 + 3] = m;
        __syncthreads();
        #pragma unroll
        for (int c = 0; c < 8; ++c) {          // 8 chunks of 4 pixels per wave
            int p0 = wv * 32 + c * 4 + 2 * half;
            float t0 = tile4[p0 * 4 + (mrow & 3)];
            float t1 = tile4[(p0 + 1) * 4 + (mrow & 3)];
            v2f a;
            a.x = (mrow < 4) ? t0 : 0.f;
            a.y = (mrow < 4) ? t1 : 0.f;
#if defined(__HIP_DEVICE_COMPILE__)
            acc = __builtin_amdgcn_wmma_f32_16x16x4_f32(
                false, a, false, a, (short)0, acc, false, false);
#endif
        }
        __syncthreads();
    }
    // D[m][n], m = vgpr index (lanes 0-15), n = lane; need 4x4 corner.
    if (lane < 4) {
        wsum[wv * 16 + 0 * 4 + lane] = acc[0];
        wsum[wv * 16 + 1 * 4 + lane] = acc[1];
        wsum[wv * 16 + 2 * 4 + lane] = acc[2];
        wsum[wv * 16 + 3 * 4 + lane] = acc[3];
    }
    __syncthreads();
    if (tid < 16) {                    // fixed-order cross-wave sum (deterministic)
        float s = 0.f;
        #pragma unroll
        for (int w = 0; w < 8; ++w) s += wsum[w * 16 + tid];
        partials[blockIdx.x * 16 + tid] = s;
    }
}

// ---------------- 3x3 symmetric eigensolver helpers (fp64, 1 thread) -------
__device__ void cross3(const double* a, const double* b, double* c) {
    c[0] = a[1] * b[2] - a[2] * b[1];
    c[1] = a[2] * b[0] - a[0] * b[2];
    c[2] = a[0] * b[1] - a[1] * b[0];
}
__device__ void eigvec_of(const double cov[3][3], double l, double v[3]) {
    double r0[3] = { cov[0][0] - l, cov[0][1], cov[0][2] };
    double r1[3] = { cov[0][1], cov[1][1] - l, cov[1][2] };
    double r2[3] = { cov[0][2], cov[1][2], cov[2][2] - l };
    double c01[3], c02[3], c12[3];
    cross3(r0, r1, c01); cross3(r0, r2, c02); cross3(r1, r2, c12);
    double n01 = c01[0]*c01[0] + c01[1]*c01[1] + c01[2]*c01[2];
    double n02 = c02[0]*c02[0] + c02[1]*c02[1] + c02[2]*c02[2];
    double n12 = c12[0]*c12[0] + c12[1]*c12[1] + c12[2]*c12[2];
    double* best = c01; double bn = n01;
    if (n02 > bn) { best = c02; bn = n02; }
    if (n12 > bn) { best = c12; bn = n12; }
    double inv = 1.0 / sqrt(bn > 1e-300 ? bn : 1.0);
    v[0] = best[0] * inv; v[1] = best[1] * inv; v[2] = best[2] * inv;
    int bi = 0; double ba = fabs(v[0]);
    if (fabs(v[1]) > ba) { bi = 1; ba = fabs(v[1]); }
    if (fabs(v[2]) > ba) { bi = 2; }
    if (v[bi] < 0.0) { v[0] = -v[0]; v[1] = -v[1]; v[2] = -v[2]; }
}

// ---------------- pass 2: reduce partials, cov, eig, store E ----------------
__global__ __launch_bounds__(256) void mk_finalize(
        const float* __restrict__ partials, float* __restrict__ sc) {
    __shared__ double S[16];
    int tid = threadIdx.x;
    if (tid < 16) {
        double s = 0.0;
        for (int b = 0; b < GRID1; ++b) s += (double)partials[b * 16 + tid];
        S[tid] = s;
    }
    __syncthreads();
    if (tid == 0) {
        double n = S[15];
        double mean[3] = { S[12] / n, S[13] / n, S[14] / n };
        double cov[3][3];
        for (int i = 0; i < 3; ++i)
            for (int j = 0; j < 3; ++j)
                cov[i][j] = (S[i * 4 + j] - n * mean[i] * mean[j]) / (n - 1.0);
        double p1 = cov[0][1]*cov[0][1] + cov[0][2]*cov[0][2] + cov[1][2]*cov[1][2];
        double qd = (cov[0][0] + cov[1][1] + cov[2][2]) / 3.0;
        double p2 = (cov[0][0]-qd)*(cov[0][0]-qd) + (cov[1][1]-qd)*(cov[1][1]-qd)
                  + (cov[2][2]-qd)*(cov[2][2]-qd) + 2.0 * p1;
        double lmax, lmid;
        if (p2 <= 1e-30) { lmax = lmid = qd; }
        else {
            double pd = sqrt(p2 / 6.0), inv = 1.0 / pd;
            double B[3][3];
            for (int i = 0; i < 3; ++i)
                for (int j = 0; j < 3; ++j)
                    B[i][j] = (cov[i][j] - (i == j ? qd : 0.0)) * inv;
            double detB = B[0][0]*(B[1][1]*B[2][2] - B[1][2]*B[1][2])
                        - B[0][1]*(B[0][1]*B[2][2] - B[1][2]*B[0][2])
                        + B[0][2]*(B[0][1]*B[1][2] - B[1][1]*B[0][2]);
            double r = detB * 0.5; r = fmin(1.0, fmax(-1.0, r));
            double phi = acos(r) / 3.0;
            double e1 = qd + 2.0 * pd * cos(phi);                       // largest
            double e3 = qd + 2.0 * pd * cos(phi + 2.0943951023931953);  // smallest
            lmax = e1; lmid = 3.0 * qd - e1 - e3;
        }
        double vmid[3], vmx[3];
        eigvec_of(cov, lmid, vmid);
        eigvec_of(cov, lmax, vmx);
        sc[0] = (float)vmid[0]; sc[1] = (float)vmx[0];   // E row-major 3x2
        sc[2] = (float)vmid[1]; sc[3] = (float)vmx[1];
        sc[4] = (float)vmid[2]; sc[5] = (float)vmx[2];
        sc[6] = (float)n;
    }
}

// ---------------- pass 3: masked phi histogram ----------------
__global__ __launch_bounds__(256) void mk_phi_hist(
        const float* __restrict__ od, const float* __restrict__ sc,
        unsigned* __restrict__ histPhi, int N) {
    int i0 = blockIdx.x * blockDim.x + threadIdx.x;
    int stride = gridDim.x * blockDim.x;
    float e00 = sc[0], e01 = sc[1], e10 = sc[2], e11 = sc[3], e20 = sc[4], e21 = sc[5];
    const float scale = (float)NB / (2.f * PI_F);
    for (int p = i0; p < N; p += stride) {
        float o0 = od[p], o1 = od[N + p], o2 = od[2 * N + p];
        if (o0 >= BETA_V && o1 >= BETA_V && o2 >= BETA_V) {
            float t0 = o0 * e00 + o1 * e10 + o2 * e20;
            float t1 = o0 * e01 + o1 * e11 + o2 * e21;
            float ph = atan2f(t1, t0);
            int b = (int)((ph + PI_F) * scale);
            b = min(max(b, 0), NB - 1);
            atomicAdd(&histPhi[b], 1u);
        }
    }
}

// ---------------- block-cooperative histogram quantile ----------------
__device__ float hist_quantile(const unsigned* __restrict__ hist,
                               float lo, float hi, float q) {
    __shared__ unsigned csum[256];
    __shared__ float res;
    const int tid = threadIdx.x;
    const int chunk = NB / 256;
    __syncthreads();
    unsigned s = 0;
    for (int i = 0; i < chunk; ++i) s += hist[tid * chunk + i];
    csum[tid] = s;
    __syncthreads();
    if (tid == 0) {
        unsigned long long tot = 0;
        for (int i = 0; i < 256; ++i) tot += csum[i];
        if (tot == 0) { res = lo; }
        else {
            double target = (double)q * (double)(tot - 1);
            unsigned long long cum = 0;
            int ci = 0;
            while (ci < 255 && (double)(cum + csum[ci]) <= target) { cum += csum[ci]; ++ci; }
            int bi = ci * chunk, last = bi + chunk;
            unsigned cnt = 0;
            while (bi < last) {
                cnt = hist[bi];
                if ((double)(cum + cnt) > target) break;
                cum += cnt; ++bi;
            }
            if (bi == last) { bi = last - 1; cnt = hist[bi] ? hist[bi] : 1u; }
            double w = (double)(hi - lo) / (double)NB;
            double frac = (target - (double)cum + 0.5) / (double)(cnt ? cnt : 1u);
            frac = fmin(1.0, fmax(0.0, frac));
            res = (float)((double)lo + w * ((double)bi + frac));
        }
    }
    __syncthreads();
    return res;
}

// ---------------- pass 4: phi quantiles -> HE -> M = (HE^T HE)^-1 HE^T ------
__global__ __launch_bounds__(256) void mk_phi_quant(
        const unsigned* __restrict__ histPhi, float* __restrict__ sc) {
    float minPhi = hist_quantile(histPhi, -PI_F, PI_F, 0.01f);
    float maxPhi = hist_quantile(histPhi, -PI_F, PI_F, 0.99f);
    if (threadIdx.x == 0) {
        float E[3][2] = { { sc[0], sc[1] }, { sc[2], sc[3] }, { sc[4], sc[5] } };
        float cmn = cosf(minPhi), smn = sinf(minPhi);
        float cmx = cosf(maxPhi), smx = sinf(maxPhi);
        float vMin[3], vMax[3], HE[3][2];
        for (int c = 0; c < 3; ++c) {
            vMin[c] = E[c][0] * cmn + E[c][1] * smn;
            vMax[c] = E[c][0] * cmx + E[c][1] * smx;
        }
        if (vMin[0] > vMax[0])
            for (int c = 0; c < 3; ++c) { HE[c][0] = vMin[c]; HE[c][1] = vMax[c]; }
        else
            for (int c = 0; c < 3; ++c) { HE[c][0] = vMax[c]; HE[c][1] = vMin[c]; }
        float g00 = 0.f, g01 = 0.f, g11 = 0.f;
        for (int c = 0; c < 3; ++c) {
            g00 += HE[c][0] * HE[c][0];
            g01 += HE[c][0] * HE[c][1];
            g11 += HE[c][1] * HE[c][1];
        }
        float inv = 1.f / (g00 * g11 - g01 * g01);
        for (int c = 0; c < 3; ++c) {
            sc[8 + c]  = ( g11 * HE[c][0] - g01 * HE[c][1]) * inv;   // M row 0
            sc[11 + c] = (-g01 * HE[c][0] + g00 * HE[c][1]) * inv;   // M row 1
        }
    }
}

// ---------------- pass 5: C min/max (ordered-uint keys, deterministic) ------
__global__ __launch_bounds__(256) void mk_c_minmax(
        const float* __restrict__ od, const float* __restrict__ sc,
        unsigned* __restrict__ mm, int N) {
    __shared__ unsigned smn0[256], smx0[256], smn1[256], smx1[256];
    int tid = threadIdx.x;
    int i0 = blockIdx.x * blockDim.x + tid;
    int stride = gridDim.x * blockDim.x;
    float m00 = sc[8], m01 = sc[9], m02 = sc[10];
    float m10 = sc[11], m11 = sc[12], m12 = sc[13];
    unsigned kn0 = 0xFFFFFFFFu, kx0 = 0u, kn1 = 0xFFFFFFFFu, kx1 = 0u;
    for (int p = i0; p < N; p += stride) {
        float o0 = od[p], o1 = od[N + p], o2 = od[2 * N + p];
        unsigned k0 = fkey(m00 * o0 + m01 * o1 + m02 * o2);
        unsigned k1 = fkey(m10 * o0 + m11 * o1 + m12 * o2);
        kn0 = min(kn0, k0); kx0 = max(kx0, k0);
        kn1 = min(kn1, k1); kx1 = max(kx1, k1);
    }
    smn0[tid] = kn0; smx0[tid] = kx0; smn1[tid] = kn1; smx1[tid] = kx1;
    __syncthreads();
    for (int o = 128; o > 0; o >>= 1) {
        if (tid < o) {
            smn0[tid] = min(smn0[tid], smn0[tid + o]);
            smx0[tid] = max(smx0[tid], smx0[tid + o]);
            smn1[tid] = min(smn1[tid], smn1[tid + o]);
            smx1[tid] = max(smx1[tid], smx1[tid + o]);
        }
        __syncthreads();
    }
    if (tid == 0) {
        atomicMin(&mm[0], smn0[0]); atomicMax(&mm[1], smx0[0]);
        atomicMin(&mm[2], smn1[0]); atomicMax(&mm[3], smx1[0]);
    }
}

// ---------------- pass 6: C histograms ----------------
__global__ __launch_bounds__(256) void mk_c_hist(
        const float* __restrict__ od, const float* __restrict__ sc,
        const unsigned* __restrict__ mm,
        unsigned* __restrict__ histC0, unsigned* __restrict__ histC1, int N) {
    int i0 = blockIdx.x * blockDim.x + threadIdx.x;
    int stride = gridDim.x * blockDim.x;
    float m00 = sc[8], m01 = sc[9], m02 = sc[10];
    float m10 = sc[11], m11 = sc[12], m12 = sc[13];
    float lo0 = fdec(mm[0]), hi0 = fdec(mm[1]);
    float lo1 = fdec(mm[2]), hi1 = fdec(mm[3]);
    float w0 = (float)NB / fmaxf(hi0 - lo0, 1e-30f);
    float w1 = (float)NB / fmaxf(hi1 - lo1, 1e-30f);
    for (int p = i0; p < N; p += stride) {
        float o0 = od[p], o1 = od[N + p], o2 = od[2 * N + p];
        float c0 = m00 * o0 + m01 * o1 + m02 * o2;
        float c1 = m10 * o0 + m11 * o1 + m12 * o2;
        int b0 = min(max((int)((c0 - lo0) * w0), 0), NB - 1);
        int b1 = min(max((int)((c1 - lo1) * w1), 0), NB - 1);
        atomicAdd(&histC0[b0], 1u);
        atomicAdd(&histC1[b1], 1u);
    }
}

// ---------------- pass 7: C quantiles -> scales ----------------
__global__ __launch_bounds__(256) void mk_c_quant(
        const unsigned* __restrict__ histC0, const unsigned* __restrict__ histC1,
        const unsigned* __restrict__ mm, const float* __restrict__ maxCRef,
        float* __restrict__ sc) {
    float lo0 = fdec(mm[0]), hi0 = fdec(mm[1]);
    float lo1 = fdec(mm[2]), hi1 = fdec(mm[3]);
    float q0 = hist_quantile(histC0, lo0, hi0, 0.99f);
    float q1 = hist_quantile(histC1, lo1, hi1, 0.99f);
    if (threadIdx.x == 0) {
        sc[14] = maxCRef[0] / q0;
        sc[15] = maxCRef[1] / q1;
    }
}

// ---------------- pass 8: output ----------------
__global__ __launch_bounds__(256) void mk_output(
        const float* __restrict__ od, const float* __restrict__ sc,
        const float* __restrict__ heref, float* __restrict__ out, int N) {
    int i0 = blockIdx.x * blockDim.x + threadIdx.x;
    int stride = gridDim.x * blockDim.x;
    float m00 = sc[8], m01 = sc[9], m02 = sc[10];
    float m10 = sc[11], m11 = sc[12], m12 = sc[13];
    float s0 = sc[14], s1 = sc[15];
    float h00 = heref[0], h01 = heref[1];
    float h10 = heref[2], h11 = heref[3];
    float h20 = heref[4], h21 = heref[5];
    const float inv255 = 1.f / 255.f;
    for (int p = i0; p < N; p += stride) {
        float o0 = od[p], o1 = od[N + p], o2 = od[2 * N + p];
        float c0 = (m00 * o0 + m01 * o1 + m02 * o2) * s0;
        float c1 = (m10 * o0 + m11 * o1 + m12 * o2) * s1;
        float r = floorf(fminf(IO_VAL * expf(-(h00 * c0 + h01 * c1)), 255.f)) * inv255;
        float g = floorf(fminf(IO_VAL * expf(-(h10 * c0 + h11 * c1)), 255.f)) * inv255;
        float b = floorf(fminf(IO_VAL * expf(-(h20 * c0 + h21 * c1)), 255.f)) * inv255;
        out[3 * p + 0] = r;
        out[3 * p + 1] = g;
        out[3 * p + 2] = b;
    }
}

// ---------------- launcher ----------------
extern "C" void kernel_launch(void* const* d_in, const int* in_sizes, int n_in,
                              void* d_out, int out_size, void* d_ws, size_t ws_size,
                              hipStream_t stream) {
    const float* img     = (const float*)d_in[0];   // (3, H, W)
    const float* heref   = (const float*)d_in[1];   // (3, 2)
    const float* maxCRef = (const float*)d_in[2];   // (2,)
    const int N = in_sizes[0] / 3;

    // workspace layout
    float*    ws_f     = (float*)d_ws;
    float*    od       = ws_f;                                  // 3N floats
    unsigned* histPhi  = (unsigned*)(ws_f + (size_t)3 * N);     // NB
    unsigned* histC0   = histPhi + NB;                          // NB
    unsigned* histC1   = histC0 + NB;                           // NB
    float*    partials = (float*)(histC1 + NB);                 // GRID1*16
    float*    sc       = partials + (size_t)GRID1 * 16;         // 64 floats
    unsigned* mm       = (unsigned*)(sc + 64);                  // 4 u32

    const int perBlock = (((N + GRID1 - 1) / GRID1) + 255) & ~255;
    const int nHist = 3 * NB;
    const int NBLK = 2048;

    mk_init     <<<(nHist + 255) / 256, 256, 0, stream>>>(histPhi, mm, nHist);
    mk_od_stats <<<GRID1, 256, 0, stream>>>(img, od, partials, N, perBlock);
    mk_finalize <<<1, 256, 0, stream>>>(partials, sc);
    mk_phi_hist <<<NBLK, 256, 0, stream>>>(od, sc, histPhi, N);
    mk_phi_quant<<<1, 256, 0, stream>>>(histPhi, sc);
    mk_c_minmax <<<NBLK, 256, 0, stream>>>(od, sc, mm, N);
    mk_c_hist   <<<NBLK, 256, 0, stream>>>(od, sc, mm, histC0, histC1, N);
    mk_c_quant  <<<1, 256, 0, stream>>>(histC0, histC1, mm, maxCRef, sc);
    mk_output   <<<NBLK, 256, 0, stream>>>(od, sc, heref, (float*)d_out, N);
}